// WalshHadamard_11879879543391
// MI455X (gfx1250) — compile-verified
//
#include <hip/hip_runtime.h>

// ---------------------------------------------------------------------------
// Walsh–Hadamard transform, N = 2^25 fp32, MI455X (gfx1250, wave32).
//
// Memory-bound (N*log2N adds vs 256MB per global pass at 23.3 TB/s ~= 11us).
// CDNA5 allows 320KB LDS per workgroup -> 256KB (2^16 float) tile, so the
// whole transform needs only TWO global passes (16 bits + 9 bits):
//   pass 1: bits 0..15  = 4 x H16 WMMA digit stages (no scalar butterflies)
//   pass 2: bits 16..24 = 2 x H16 WMMA digits + 1 butterfly, 64KB tile
// Each H16 digit = D = H16 * X via four chained V_WMMA_F32_16X16X4_F32
// (full fp32; H16 entries +/-1 generated in-register from popcount(M&K)).
// Global<->LDS movement uses CDNA5 async-to-LDS ops (ASYNCcnt), skipping the
// VGPR round-trip (GLOBAL_LOAD_ASYNC_TO_LDS_B128 / STORE_ASYNC_FROM_LDS).
// ---------------------------------------------------------------------------

typedef __attribute__((ext_vector_type(2))) float v2f;
typedef __attribute__((ext_vector_type(8))) float v8f;

#define NTHREADS 256        // 8 wave32 waves per workgroup
#define TILE1    65536      // pass 1 tile: 256 KB LDS (ISA max: 320 KB / WG)
#define TILE2    16384      // pass 2 tile:  64 KB LDS

// ---- async global<->LDS path (probe-guarded) ------------------------------
#if defined(__has_builtin)
#if __has_builtin(__builtin_amdgcn_global_load_async_to_lds_b128) && \
    __has_builtin(__builtin_amdgcn_global_store_async_from_lds_b128) && \
    __has_builtin(__builtin_amdgcn_s_wait_asynccnt)
#define HAVE_ASYNC_LDS 1
#endif
#endif
#ifndef HAVE_ASYNC_LDS
#define HAVE_ASYNC_LDS 0
#endif

#if HAVE_ASYNC_LDS
// Builtin parameter type (per hipcc diagnostic): pointer to
// 'int __attribute__((vector_size(16)))' in AS1 (global) / AS3 (LDS).
typedef int v4i_vs __attribute__((vector_size(16)));
typedef __attribute__((address_space(1))) v4i_vs* gptr_t;
typedef __attribute__((address_space(3))) v4i_vs* lptr_t;
// Generic->AS casts via integer punning (avoids addrspacecast diagnostics).
// Flat LDS address = aperture | offset; LDS hardware uses addr[31:0].
__device__ __forceinline__ gptr_t as_global(const void* p) {
  return (gptr_t)(unsigned long long)p;
}
__device__ __forceinline__ lptr_t as_lds(const void* p) {
  return (lptr_t)(unsigned)(unsigned long long)p;
}
#endif

// Copy nVec float4's  global<->LDS, strided by all NTHREADS threads.
__device__ __forceinline__ void tile_load(float4* lds4, const float4* g4,
                                          int nVec, int tid) {
#if HAVE_ASYNC_LDS
  for (int i = tid; i < nVec; i += NTHREADS)
    __builtin_amdgcn_global_load_async_to_lds_b128(as_global(g4 + i),
                                                   as_lds(lds4 + i), 0, 0);
  __builtin_amdgcn_s_wait_asynccnt(0);
#else
  for (int i = tid; i < nVec; i += NTHREADS) lds4[i] = g4[i];
#endif
}
__device__ __forceinline__ void tile_store(float4* g4, const float4* lds4,
                                           int nVec, int tid) {
#if HAVE_ASYNC_LDS
  for (int i = tid; i < nVec; i += NTHREADS)
    __builtin_amdgcn_global_store_async_from_lds_b128(as_global(g4 + i),
                                                      as_lds(lds4 + i), 0, 0);
  // S_ENDPGM performs an implicit wait-idle; no explicit asynccnt wait needed.
#else
  for (int i = tid; i < nVec; i += NTHREADS) g4[i] = lds4[i];
#endif
}

// ---- H16 digit via WMMA ---------------------------------------------------
// Apply D = H16 * X to one 16x16 tile in LDS.
//   colBase : per-lane LDS float offset of this lane's column (N = lane & 15)
//   S       : LDS element stride along the 16-element (K / M) axis
// Layouts (CDNA5 ISA 7.12.2, 32-bit operands):
//   A 16x4 : lanes 0-15 -> M=lane, VGPR{0,1}=K{0,1}; lanes 16-31 -> K{2,3}
//   B 4x16 : lanes 0-15 -> N=lane, VGPR{0,1}=K{0,1}; lanes 16-31 -> K{2,3}
//   C/D    : lane N = lane&15, VGPR v -> M = v + 8*(lane>=16)
__device__ __forceinline__ void h16_tile(float* lds, int colBase, int S, int lane) {
  const int kB = (lane >> 4) << 1;   // this lane's K sub-slots: {0,1} or {2,3}
  const int m  = lane & 15;          // A-matrix row
  v8f acc = {};
  #pragma unroll
  for (int c = 0; c < 4; ++c) {      // K chunks 0..3 -> full K=16 reduction
    const int k0 = 4 * c + kB;
    const int k1 = k0 + 1;
    v2f a, b;
    // H16[m][k] = (-1)^popcount(m & k)  (exact +/-1 in fp32)
    a.x = (__popc((unsigned)(m & k0)) & 1) ? -1.0f : 1.0f;
    a.y = (__popc((unsigned)(m & k1)) & 1) ? -1.0f : 1.0f;
    b.x = lds[colBase + k0 * S];
    b.y = lds[colBase + k1 * S];
    // (neg_a, A, neg_b, B, c_mod, C, reuse_a, reuse_b)
    acc = __builtin_amdgcn_wmma_f32_16x16x4_f32(
        false, a, false, b, (short)0, acc, false, false);
  }
  const int mBase = (lane >> 4) << 3;  // this lane writes rows 0-7 or 8-15
  #pragma unroll
  for (int v = 0; v < 8; ++v)
    lds[colBase + (mBase + v) * S] = acc[v];
}

// One H16 digit stage over a T-float tile: transforms flat-index bits
// [log2 S, log2 S + 4).  Group loop is wave-uniform (readfirstlane) so WMMA
// always runs with EXEC all-ones under scalar loop control.
template <int T>
__device__ __forceinline__ void digit_stage(float* lds, int tid, int S) {
  const int lane = tid & 31;
  const int wave = __builtin_amdgcn_readfirstlane(tid >> 5);
  for (int g = wave; g < T / 256; g += NTHREADS / 32) {
    const int cc   = (g << 4) + (lane & 15);                  // column id
    const int base = ((cc & ~(S - 1)) << 4) | (cc & (S - 1)); // insert digit
    h16_tile(lds, base, S, lane);
  }
}

// One classic H2 butterfly stage on flat-index bit log2(S).
template <int T>
__device__ __forceinline__ void bfly_stage(float* lds, int tid, int S) {
  for (int i = tid; i < T / 2; i += NTHREADS) {
    const int lo = ((i & ~(S - 1)) << 1) | (i & (S - 1));
    const int hi = lo + S;
    const float a = lds[lo];
    const float b = lds[hi];
    lds[lo] = a + b;
    lds[hi] = a - b;
  }
}

// Pass 1: global bits 0..15.  Contiguous 64K-float tile per block,
// four pure-WMMA digit stages.  512 blocks.
__global__ __launch_bounds__(NTHREADS)
void wht_pass_low(const float* __restrict__ x, float* __restrict__ y) {
  __shared__ __align__(16) float tile[TILE1];
  const int tid = threadIdx.x;
  float4* t4 = (float4*)tile;
  tile_load(t4, (const float4*)(x + (size_t)blockIdx.x * TILE1), TILE1 / 4, tid);
  __syncthreads();
  digit_stage<TILE1>(tile, tid, 1);     __syncthreads();   // bits  0-3
  digit_stage<TILE1>(tile, tid, 16);    __syncthreads();   // bits  4-7
  digit_stage<TILE1>(tile, tid, 256);   __syncthreads();   // bits  8-11
  digit_stage<TILE1>(tile, tid, 4096);  __syncthreads();   // bits 12-15
  tile_store((float4*)(y + (size_t)blockIdx.x * TILE1), t4, TILE1 / 4, tid);
}

// Pass 2: global bits 16..24.  Tile = 512 rows (stride 2^16) x 32 contiguous
// columns; 8 lanes x float4 = one 128B row segment (fully coalesced).
// blockIdx covers index bits 5..15.  2048 blocks.
__global__ __launch_bounds__(NTHREADS)
void wht_pass_high(float* __restrict__ y) {
  __shared__ __align__(16) float tile[TILE2];
  const int tid = threadIdx.x;
  const size_t blockBase = (size_t)blockIdx.x * 32;   // index bits 5..15
  float4* t4 = (float4*)tile;
  for (int i = tid; i < TILE2 / 4; i += NTHREADS) {   // i = r*8 + c4
    const int r = i >> 3, c4 = i & 7;
    const float4* row = (const float4*)(y + blockBase + ((size_t)r << 16));
#if HAVE_ASYNC_LDS
    __builtin_amdgcn_global_load_async_to_lds_b128(as_global(row + c4),
                                                   as_lds(t4 + i), 0, 0);
#else
    t4[i] = row[c4];
#endif
  }
#if HAVE_ASYNC_LDS
  __builtin_amdgcn_s_wait_asynccnt(0);
#endif
  __syncthreads();
  digit_stage<TILE2>(tile, tid, 32);   __syncthreads();   // global bits 16-19
  digit_stage<TILE2>(tile, tid, 512);  __syncthreads();   // global bits 20-23
  bfly_stage<TILE2>(tile, tid, 8192);  __syncthreads();   // global bit  24
  for (int i = tid; i < TILE2 / 4; i += NTHREADS) {
    const int r = i >> 3, c4 = i & 7;
    float4* row = (float4*)(y + blockBase + ((size_t)r << 16));
#if HAVE_ASYNC_LDS
    __builtin_amdgcn_global_store_async_from_lds_b128(as_global(row + c4),
                                                      as_lds(t4 + i), 0, 0);
#else
    row[c4] = t4[i];
#endif
  }
}

extern "C" void kernel_launch(void* const* d_in, const int* in_sizes, int n_in,
                              void* d_out, int out_size, void* d_ws, size_t ws_size,
                              hipStream_t stream) {
  const float* x = (const float*)d_in[0];   // 2^25 fp32 (h_dim_exp == 25)
  float* y = (float*)d_out;
  wht_pass_low <<<out_size / TILE1, NTHREADS, 0, stream>>>(x, y);  //  512 blocks
  wht_pass_high<<<out_size / TILE2, NTHREADS, 0, stream>>>(y);     // 2048 blocks
}